// Generator_12833362280664
// MI455X (gfx1250) — compile-verified
//
#include <hip/hip_runtime.h>
#include <math.h>

#define NREG   4096
#define EDGES  131072
#define FIN    131
#define KPAD   144      // FIN padded to a multiple of 16 (zero-filled)
#define HEADS  6
#define DDIM   64
#define HD     384      // HEADS*DDIM
#define BWALK  256
#define WSTEPS 199

typedef float v2f __attribute__((ext_vector_type(2)));
typedef float v8f __attribute__((ext_vector_type(8)));

// ---------------------------------------------------------------------------
// Zero-padding for layer 0 so the GEMM inner loop needs no bound guards.
// ---------------------------------------------------------------------------
__global__ void pad_x_kernel(const float* __restrict__ x, float* __restrict__ xp) {
  int i = blockIdx.x * blockDim.x + threadIdx.x;          // over NREG*KPAD
  if (i >= NREG * KPAD) return;
  int n = i / KPAD, k = i % KPAD;
  xp[i] = (k < FIN) ? x[n * FIN + k] : 0.f;
}
__global__ void pad_w_kernel(const float* __restrict__ w, float* __restrict__ wp) {
  int i = blockIdx.x * blockDim.x + threadIdx.x;          // over KPAD*HD
  if (i >= KPAD * HD) return;
  int k = i / HD, c = i % HD;
  wp[i] = (k < FIN) ? w[k * HD + c] : 0.f;
}

// ---------------------------------------------------------------------------
// GEMM: C[4096, 384] = A[4096, K] * B[K, 384], fp32 WMMA 16x16x4.
// Requires K % 16 == 0 (144 or 384). One wave per 16x16 tile; 6144 tiles,
// 8 waves/block -> 768 blocks. No guards -> no exec manipulation around WMMA.
// A frag (16x4): lane<16 -> M=lane, K=k0,k0+1 ; lane>=16 -> K=k0+2,k0+3 (b64 load).
// B frag (4x16): VGPR0 = row k0 (lanes 0-15) / k0+2 (lanes 16-31), VGPR1 = k0+1/k0+3.
// C/D: VGPR r -> M = r (lanes 0-15) or r+8 (lanes 16-31), N = lane&15.
// ---------------------------------------------------------------------------
__global__ __launch_bounds__(256)
void gemm_wmma_f32(const float* __restrict__ A, const float* __restrict__ B,
                   float* __restrict__ C, int K, int lda) {
  const int ldb = HD, ldc = HD;
  int wave = threadIdx.x >> 5;
  int lane = threadIdx.x & 31;
  int tile = blockIdx.x * 8 + wave;          // 0 .. 6143
  int tm = tile / (HD / 16);
  int tn = tile % (HD / 16);
  int half = lane >> 4;
  int l15  = lane & 15;
  const float* __restrict__ Arow = A + (tm * 16 + l15) * lda + 2 * half;
  const float* __restrict__ Bcol = B + tn * 16 + l15 + 2 * half * ldb;
  v8f acc = {};
  for (int k0 = 0; k0 < K; k0 += 16) {
    v2f a[4], b[4];
#pragma unroll
    for (int u = 0; u < 4; ++u) {
      int kk = k0 + 4 * u;
      a[u] = *(const v2f*)(Arow + kk);          // global_load_b64, contiguous
      b[u].x = Bcol[kk * ldb];
      b[u].y = Bcol[(kk + 1) * ldb];
    }
#pragma unroll
    for (int u = 0; u < 4; ++u)
      acc = __builtin_amdgcn_wmma_f32_16x16x4_f32(false, a[u], false, b[u],
                                                  (short)0, acc, false, false);
  }
#pragma unroll
  for (int r = 0; r < 8; ++r)
    C[(tm * 16 + r + 8 * half) * ldc + tn * 16 + l15] = acc[r];
}

// ---------------------------------------------------------------------------
// el/er: per (node, head) dot of feat[n,h,:] with al[h,:], ar[h,:]
// ---------------------------------------------------------------------------
__global__ void elr_kernel(const float* __restrict__ feat,
                           const float* __restrict__ al,
                           const float* __restrict__ ar,
                           float* __restrict__ el, float* __restrict__ er) {
  int idx = blockIdx.x * blockDim.x + threadIdx.x;
  if (idx >= NREG * HEADS) return;
  int n = idx / HEADS, h = idx % HEADS;
  const float* f = feat + n * HD + h * DDIM;
  const float* a = al + h * DDIM;
  const float* r = ar + h * DDIM;
  float sl = 0.f, sr = 0.f;
  for (int d = 0; d < DDIM; ++d) { sl += f[d] * a[d]; sr += f[d] * r[d]; }
  el[idx] = sl; er[idx] = sr;
}

// e = leaky_relu(el[src] + er[dst], 0.2) per (edge, head)
__global__ void edge_e_kernel(const int* __restrict__ src, const int* __restrict__ dst,
                              const float* __restrict__ el, const float* __restrict__ er,
                              float* __restrict__ ebuf) {
  int idx = blockIdx.x * blockDim.x + threadIdx.x;
  if (idx >= EDGES * HEADS) return;
  int e = idx / HEADS, h = idx % HEADS;
  float v = el[src[e] * HEADS + h] + er[dst[e] * HEADS + h];
  ebuf[idx] = v > 0.f ? v : 0.2f * v;
}

// per (dst, head): stable softmax over incoming edges; ebuf <- exp(e-m), den <- sum
__global__ void edge_softmax_kernel(const int* __restrict__ offs, const int* __restrict__ csre,
                                    float* __restrict__ ebuf, float* __restrict__ den) {
  int idx = blockIdx.x * blockDim.x + threadIdx.x;
  if (idx >= NREG * HEADS) return;
  int n = idx / HEADS, h = idx % HEADS;
  int s = offs[n], t = offs[n + 1];
  float m = -INFINITY;
  for (int j = s; j < t; ++j) m = fmaxf(m, ebuf[csre[j] * HEADS + h]);
  if (!(m > -INFINITY)) m = 0.f;   // zero in-degree: jnp.where(isfinite, m, 0)
  float sum = 0.f;
  for (int j = s; j < t; ++j) {
    int e = csre[j];
    float ex = expf(ebuf[e * HEADS + h] - m);
    ebuf[e * HEADS + h] = ex;
    sum += ex;
  }
  den[idx] = sum;
}

// block per dst node; 384 threads = (h,d). mode 0: ELU -> [N,HD]; mode 1: head-mean -> [N,64]
__global__ __launch_bounds__(384)
void aggregate_kernel(const int* __restrict__ offs, const int* __restrict__ csre,
                      const int* __restrict__ src,
                      const float* __restrict__ feat, const float* __restrict__ ebuf,
                      const float* __restrict__ den, const float* __restrict__ bias,
                      float* __restrict__ outp, int mode) {
  __shared__ float sh[HD];
  int n = blockIdx.x;
  int t = threadIdx.x;
  int h = t / DDIM;
  int s = offs[n], e2 = offs[n + 1];
  float acc = 0.f;
  for (int j = s; j < e2; ++j) {
    int eid = csre[j];
    acc += ebuf[eid * HEADS + h] * feat[src[eid] * HD + t];
  }
  float rst = (e2 > s) ? acc / den[n * HEADS + h] : 0.f;
  rst += bias[t];
  if (mode == 0) {
    outp[n * HD + t] = rst > 0.f ? rst : (expf(rst) - 1.f);   // ELU
  } else {
    sh[t] = rst;
    __syncthreads();
    if (t < DDIM) {
      float m = 0.f;
      for (int hh = 0; hh < HEADS; ++hh) m += sh[hh * DDIM + t];
      outp[n * DDIM + t] = m * (1.f / HEADS);
    }
  }
}

// a[n] = emb[n].w_lp[0:64], c[n] = emb[n].w_lp[64:128]
__global__ void ac_kernel(const float* __restrict__ emb, const float* __restrict__ wlp,
                          float* __restrict__ aw, float* __restrict__ cw) {
  int n = blockIdx.x * blockDim.x + threadIdx.x;
  if (n >= NREG) return;
  const float* e = emb + n * DDIM;
  float sa = 0.f, sc = 0.f;
  for (int d = 0; d < DDIM; ++d) { sa += e[d] * wlp[d]; sc += e[d] * wlp[DDIM + d]; }
  aw[n] = sa; cw[n] = sc;
}

// S[i] = sum_j tanh(a[j] + c[i] + dist[i,j]*w + b)
__global__ __launch_bounds__(256)
void rowsum_kernel(const float* __restrict__ aw, const float* __restrict__ cw,
                   const float* __restrict__ dist, const float* __restrict__ wlp,
                   const float* __restrict__ blp, float* __restrict__ S) {
  __shared__ float red[256];
  int i = blockIdx.x;
  float ci = cw[i], wd = wlp[2 * DDIM], bb = blp[0];
  float sum = 0.f;
  for (int j = threadIdx.x; j < NREG; j += 256)
    sum += tanhf(aw[j] + ci + dist[(size_t)i * NREG + j] * wd + bb);
  red[threadIdx.x] = sum; __syncthreads();
  for (int st = 128; st > 0; st >>= 1) {
    if (threadIdx.x < st) red[threadIdx.x] += red[threadIdx.x + st];
    __syncthreads();
  }
  if (threadIdx.x == 0) S[i] = red[0];
}

// ------------------------- CSR construction --------------------------------
__global__ void zero_int_kernel(int* p, int n) {
  int i = blockIdx.x * blockDim.x + threadIdx.x;
  if (i < n) p[i] = 0;
}
__global__ void hist_kernel(const int* __restrict__ dst, int* __restrict__ cnt) {
  int e = blockIdx.x * blockDim.x + threadIdx.x;
  if (e < EDGES) atomicAdd(&cnt[dst[e]], 1);
}
__global__ __launch_bounds__(1024)
void scan_kernel(const int* __restrict__ cnt, int* __restrict__ offs) {
  __shared__ int part[1024];
  int t = threadIdx.x, base = t * 4;
  int c0 = cnt[base], c1 = cnt[base + 1], c2 = cnt[base + 2], c3 = cnt[base + 3];
  int local = c0 + c1 + c2 + c3;
  part[t] = local; __syncthreads();
  for (int off = 1; off < 1024; off <<= 1) {
    int v = (t >= off) ? part[t - off] : 0;
    __syncthreads();
    part[t] += v;
    __syncthreads();
  }
  int excl = part[t] - local;
  offs[base] = excl; offs[base + 1] = excl + c0;
  offs[base + 2] = excl + c0 + c1; offs[base + 3] = excl + c0 + c1 + c2;
  if (t == 1023) offs[NREG] = part[1023];
}
__global__ void scatter_kernel(const int* __restrict__ dst, const int* __restrict__ offs,
                               int* __restrict__ cursor, int* __restrict__ csre) {
  int e = blockIdx.x * blockDim.x + threadIdx.x;
  if (e < EDGES) {
    int d = dst[e];
    int p = atomicAdd(&cursor[d], 1);
    csre[offs[d] + p] = e;
  }
}
// restore determinism: sort each bucket ascending by edge id
__global__ void sort_kernel(const int* __restrict__ offs, int* __restrict__ csre) {
  int n = blockIdx.x * blockDim.x + threadIdx.x;
  if (n >= NREG) return;
  int s = offs[n], e = offs[n + 1];
  for (int i = s + 1; i < e; ++i) {
    int v = csre[i], j = i - 1;
    while (j >= s && csre[j] > v) { csre[j + 1] = csre[j]; --j; }
    csre[j + 1] = v;
  }
}

// ------------------------- JAX threefry2x32 --------------------------------
__device__ __forceinline__ void tf2x32(unsigned k0, unsigned k1, unsigned c0, unsigned c1,
                                       unsigned& o0, unsigned& o1) {
  unsigned ks2 = k0 ^ k1 ^ 0x1BD11BDAu;
  unsigned x0 = c0 + k0, x1 = c1 + k1;
#define TF_RND(r) { x0 += x1; x1 = (x1 << (r)) | (x1 >> (32 - (r))); x1 ^= x0; }
  TF_RND(13) TF_RND(15) TF_RND(26) TF_RND(6)  x0 += k1;  x1 += ks2 + 1u;
  TF_RND(17) TF_RND(29) TF_RND(16) TF_RND(24) x0 += ks2; x1 += k0  + 2u;
  TF_RND(13) TF_RND(15) TF_RND(26) TF_RND(6)  x0 += k0;  x1 += k1  + 3u;
  TF_RND(17) TF_RND(29) TF_RND(16) TF_RND(24) x0 += k1;  x1 += ks2 + 4u;
  TF_RND(13) TF_RND(15) TF_RND(26) TF_RND(6)  x0 += ks2; x1 += k0  + 5u;
#undef TF_RND
  o0 = x0; o1 = x1;
}
// jax.random.split(key, 2): counts iota(4); blocks (0,2),(1,3);
// keys = reshape(concat(o0s,o1s),(2,2)) -> k0=(a0,b0), k1=(a1,b1)
__device__ __forceinline__ void jax_split2(unsigned k0, unsigned k1, unsigned out[4]) {
  unsigned a0, a1, b0, b1;
  tf2x32(k0, k1, 0u, 2u, a0, a1);
  tf2x32(k0, k1, 1u, 3u, b0, b1);
  out[0] = a0; out[1] = b0; out[2] = a1; out[3] = b1;
}

// ------------------------- random walks ------------------------------------
__global__ __launch_bounds__(256)
void walk_kernel(const float* __restrict__ aw, const float* __restrict__ cw,
                 const float* __restrict__ S, const float* __restrict__ dist,
                 const float* __restrict__ wlp, const float* __restrict__ blp,
                 float* __restrict__ out) {
  __shared__ float aLds[NREG];
  __shared__ float bval[256];
  __shared__ int   bidx[256];
  __shared__ int   curS;
  __shared__ unsigned keyS[2], subS[2];
  int b = blockIdx.x, t = threadIdx.x;
  for (int j = t; j < NREG; j += 256) aLds[j] = aw[j];
  float wd = wlp[2 * DDIM], bb = blp[0];
  if (t == 0) {
    // walk key b of split(key(42)=(0,42), 256): counts iota(512)
    unsigned kw[2];
    for (int q = 0; q < 2; ++q) {
      int i = 2 * b + q;
      unsigned o0, o1;
      if (i < 256) { tf2x32(0u, 42u, (unsigned)i,        (unsigned)(256 + i), o0, o1); kw[q] = o0; }
      else         { tf2x32(0u, 42u, (unsigned)(i - 256),(unsigned)i,         o0, o1); kw[q] = o1; }
    }
    unsigned sp[4]; jax_split2(kw[0], kw[1], sp);          // k_init, k_loop
    unsigned r0, r1; tf2x32(sp[0], sp[1], 0u, 0u, r0, r1); // randint bits (odd-size pad)
    curS = (int)(r0 & (NREG - 1));
    keyS[0] = sp[2]; keyS[1] = sp[3];
  }
  __syncthreads();
  const float tiny = 1.1754943508222875e-38f;
  for (int s = 0; s < WSTEPS; ++s) {
    if (t == 0) {
      unsigned sp[4]; jax_split2(keyS[0], keyS[1], sp);
      keyS[0] = sp[0]; keyS[1] = sp[1];   // kk
      subS[0] = sp[2]; subS[1] = sp[3];   // sub
    }
    __syncthreads();
    int cur = curS;
    unsigned s0 = subS[0], s1 = subS[1];
    float ccur = cw[cur];
    float Sc = S[cur] + 1e-10f;
    const float* drow = dist + (size_t)cur * NREG;
    float best = -INFINITY; int bi = NREG;
    // counts iota(4096): block p pairs lanes (p, p+2048)
    for (int p = t; p < NREG / 2; p += 256) {
      unsigned o0, o1;
      tf2x32(s0, s1, (unsigned)p, (unsigned)(p + 2048), o0, o1);
#pragma unroll
      for (int q = 0; q < 2; ++q) {
        int j = q ? p + 2048 : p;
        unsigned bits = q ? o1 : o0;
        float frac = __uint_as_float((bits >> 9) | 0x3f800000u) - 1.0f;
        float u = fmaxf(tiny, frac * (1.0f - tiny) + tiny);
        float g = -logf(-logf(u));
        float od = tanhf(aLds[j] + ccur + drow[j] * wd + bb);
        float lp = logf(od / Sc + 1e-10f) + g;
        if (lp > best || (lp == best && j < bi)) { best = lp; bi = j; }
      }
    }
    bval[t] = best; bidx[t] = bi; __syncthreads();
    for (int st = 128; st > 0; st >>= 1) {
      if (t < st) {
        float v2 = bval[t + st]; int i2 = bidx[t + st];
        if (v2 > bval[t] || (v2 == bval[t] && i2 < bidx[t])) { bval[t] = v2; bidx[t] = i2; }
      }
      __syncthreads();
    }
    if (t == 0) {
      int nxt = bidx[0];
      float od = tanhf(aLds[nxt] + ccur + drow[nxt] * wd + bb);
      out[b * WSTEPS + s] = (od > 1.0f) ? od : 0.f;   // nn.Threshold(1, 0)
      curS = nxt;
    }
    __syncthreads();
  }
}

// ---------------------------------------------------------------------------
extern "C" void kernel_launch(void* const* d_in, const int* in_sizes, int n_in,
                              void* d_out, int out_size, void* d_ws, size_t ws_size,
                              hipStream_t stream) {
  const float* nfeats = (const float*)d_in[0];
  const float* dist   = (const float*)d_in[1];
  const int*   esrc   = (const int*)d_in[2];
  const int*   edst   = (const int*)d_in[3];
  const float* W[4]  = {(const float*)d_in[4],  (const float*)d_in[8],
                        (const float*)d_in[12], (const float*)d_in[16]};
  const float* AL[4] = {(const float*)d_in[5],  (const float*)d_in[9],
                        (const float*)d_in[13], (const float*)d_in[17]};
  const float* AR[4] = {(const float*)d_in[6],  (const float*)d_in[10],
                        (const float*)d_in[14], (const float*)d_in[18]};
  const float* BB[4] = {(const float*)d_in[7],  (const float*)d_in[11],
                        (const float*)d_in[15], (const float*)d_in[19]};
  const float* wlp = (const float*)d_in[20];
  const float* blp = (const float*)d_in[21];
  float* out = (float*)d_out;

  // workspace carve (~21 MB)
  float* wsf  = (float*)d_ws;
  float* hbuf = wsf;                       // [N, HD]
  float* feat = hbuf + NREG * HD;          // [N, HD]
  float* el   = feat + NREG * HD;          // [N, H]
  float* er   = el + NREG * HEADS;         // [N, H]
  float* den  = er + NREG * HEADS;         // [N, H]
  float* ebuf = den + NREG * HEADS;        // [E, H]
  float* aw   = ebuf + (size_t)EDGES * HEADS;  // [N]
  float* cwv  = aw + NREG;                 // [N]
  float* Srow = cwv + NREG;                // [N]
  float* emb  = Srow + NREG;               // [N, 64]
  float* xpad = emb + NREG * DDIM;         // [N, KPAD]
  float* wpad = xpad + NREG * KPAD;        // [KPAD, HD]
  int* cnt  = (int*)(wpad + KPAD * HD);    // [N]
  int* offs = cnt + NREG;                  // [N+1]
  int* csre = offs + NREG + 1;             // [E]

  // --- CSR by destination (deterministic after per-bucket sort) ---
  zero_int_kernel<<<16, 256, 0, stream>>>(cnt, NREG);
  hist_kernel<<<EDGES / 256, 256, 0, stream>>>(edst, cnt);
  scan_kernel<<<1, 1024, 0, stream>>>(cnt, offs);
  zero_int_kernel<<<16, 256, 0, stream>>>(cnt, NREG);
  scatter_kernel<<<EDGES / 256, 256, 0, stream>>>(edst, offs, cnt, csre);
  sort_kernel<<<16, 256, 0, stream>>>(offs, csre);

  // --- layer-0 operand padding (exact: zero fill) ---
  pad_x_kernel<<<(NREG * KPAD + 255) / 256, 256, 0, stream>>>(nfeats, xpad);
  pad_w_kernel<<<(KPAD * HD + 255) / 256, 256, 0, stream>>>(W[0], wpad);

  // --- 4 GAT layers ---
  for (int l = 0; l < 4; ++l) {
    const float* X  = (l == 0) ? xpad : hbuf;
    const float* Wl = (l == 0) ? wpad : W[l];
    int K = (l == 0) ? KPAD : HD;
    gemm_wmma_f32<<<768, 256, 0, stream>>>(X, Wl, feat, K, K);
    elr_kernel<<<(NREG * HEADS) / 256, 256, 0, stream>>>(feat, AL[l], AR[l], el, er);
    edge_e_kernel<<<(EDGES * HEADS) / 256, 256, 0, stream>>>(esrc, edst, el, er, ebuf);
    edge_softmax_kernel<<<(NREG * HEADS) / 256, 256, 0, stream>>>(offs, csre, ebuf, den);
    aggregate_kernel<<<NREG, HD, 0, stream>>>(offs, csre, esrc, feat, ebuf, den,
                                              BB[l], (l < 3) ? hbuf : emb, (l < 3) ? 0 : 1);
  }

  // --- linear prediction pieces + row sums ---
  ac_kernel<<<NREG / 256, 256, 0, stream>>>(emb, wlp, aw, cwv);
  rowsum_kernel<<<NREG, 256, 0, stream>>>(aw, cwv, dist, wlp, blp, Srow);

  // --- 256 gumbel-argmax walks, 199 steps each ---
  walk_kernel<<<BWALK, 256, 0, stream>>>(aw, cwv, Srow, dist, wlp, blp, out);
}